// NMF3D_69226282877033
// MI455X (gfx1250) — compile-verified
//
#include <hip/hip_runtime.h>
#include <hip/hip_bf16.h>

typedef float v2f __attribute__((ext_vector_type(2)));
typedef float v8f __attribute__((ext_vector_type(8)));

#define BSN 2
#define DD  512
#define NPT 65536
#define RR  64
#define EPSF 1e-6f

static __device__ __forceinline__ v8f wmma4(v2f a, v2f b, v8f c) {
  // V_WMMA_F32_16X16X4_F32: D = A(16x4, f32) x B(4x16, f32) + C(16x16, f32)
  return __builtin_amdgcn_wmma_f32_16x16x4_f32(false, a, false, b, (short)0, c, false, false);
}

// ---------------- small VALU kernels ----------------

__global__ void k_norm_bases(const float* __restrict__ bases, float* __restrict__ Bn) {
  int bs = blockIdx.x;
  __shared__ float nrm[RR];
  for (int i = threadIdx.x; i < RR; i += blockDim.x) nrm[i] = 0.f;
  __syncthreads();
  const float* src = bases + (size_t)bs * DD * RR;
  for (int idx = threadIdx.x; idx < DD * RR; idx += blockDim.x) {
    float v = src[idx];
    atomicAdd(&nrm[idx & 63], v * v);
  }
  __syncthreads();
  float* dst = Bn + (size_t)bs * DD * RR;
  for (int idx = threadIdx.x; idx < DD * RR; idx += blockDim.x)
    dst[idx] = src[idx] / fmaxf(sqrtf(nrm[idx & 63]), 1e-12f);
}

__global__ void k_softmax(const float* __restrict__ numC, float* __restrict__ c) {
  int bs = blockIdx.y;
  int n  = blockIdx.x * blockDim.x + threadIdx.x;
  const float4* p = (const float4*)(numC + (size_t)bs * NPT * RR + (size_t)n * RR);
  float4 v[16];
  float m = -3.0e38f;
  #pragma unroll
  for (int i = 0; i < 16; i++) {
    v[i] = p[i];
    m = fmaxf(m, fmaxf(fmaxf(v[i].x, v[i].y), fmaxf(v[i].z, v[i].w)));
  }
  float s = 0.f;
  #pragma unroll
  for (int i = 0; i < 16; i++) {
    v[i].x = __expf(v[i].x - m); v[i].y = __expf(v[i].y - m);
    v[i].z = __expf(v[i].z - m); v[i].w = __expf(v[i].w - m);
    s += v[i].x + v[i].y + v[i].z + v[i].w;
  }
  float inv = 1.f / s;
  float4* o = (float4*)(c + (size_t)bs * NPT * RR + (size_t)n * RR);
  #pragma unroll
  for (int i = 0; i < 16; i++) {
    float4 w = v[i];
    w.x *= inv; w.y *= inv; w.z *= inv; w.w *= inv;
    o[i] = w;
  }
}

__global__ void k_zero(float* __restrict__ p, int n) {
  int i = blockIdx.x * blockDim.x + threadIdx.x;
  if (i < n) p[i] = 0.f;
}

__global__ void k_clamp(float* __restrict__ p, int n) {
  for (int i = blockIdx.x * blockDim.x + threadIdx.x; i < n; i += gridDim.x * blockDim.x)
    p[i] = fmaxf(p[i], EPSF);
}

__global__ void k_update_b(float* __restrict__ Bn, const float* __restrict__ numB,
                           const float* __restrict__ CtC, float* __restrict__ Btmp) {
  int bs = blockIdx.x;
  __shared__ float sC[RR * RR];
  __shared__ float nrm[RR];
  for (int i = threadIdx.x; i < RR * RR; i += blockDim.x) sC[i] = CtC[(size_t)bs * RR * RR + i];
  for (int i = threadIdx.x; i < RR; i += blockDim.x) nrm[i] = 0.f;
  __syncthreads();
  const float* bp = Bn   + (size_t)bs * DD * RR;
  const float* nb = numB + (size_t)bs * DD * RR;
  float*       bt = Btmp + (size_t)bs * DD * RR;
  for (int idx = threadIdx.x; idx < DD * RR; idx += blockDim.x) {
    int d = idx >> 6, r = idx & 63;
    float den = EPSF;
    #pragma unroll 8
    for (int s = 0; s < RR; s++) den += bp[d * RR + s] * sC[s * RR + r];
    float v = fmaxf(bp[idx] * nb[idx] / den, EPSF);
    bt[idx] = v;
    atomicAdd(&nrm[r], v * v);
  }
  __syncthreads();
  float* bo = Bn + (size_t)bs * DD * RR;
  for (int idx = threadIdx.x; idx < DD * RR; idx += blockDim.x)
    bo[idx] = bt[idx] / fmaxf(sqrtf(nrm[idx & 63]), 1e-12f);
}

// ---------------- WMMA kernels ----------------

// numC[bs][n][r] = sum_d x[bs][d][n] * Bn[bs][d][r]   (optionally clamp x)
// block = 128 (4 waves); wave -> 16 n rows x 64 r cols; grid (N/64, BS)
__global__ void k_gemm_xtb(const float* __restrict__ x, const float* __restrict__ Bn,
                           float* __restrict__ numC, int clampX) {
  int bs   = blockIdx.y;
  int lane = threadIdx.x & 31, wave = threadIdx.x >> 5;
  int l16  = lane & 15, half = lane >> 4;
  int nb   = blockIdx.x * 64 + wave * 16;
  const float* xp = x  + (size_t)bs * DD * NPT;
  const float* bp = Bn + (size_t)bs * DD * RR;
  v8f a0 = {}, a1 = {}, a2 = {}, a3 = {};
  for (int k = 0; k < DD; k += 4) {
    int ka = k + 2 * half;
    v2f a;
    float f0 = xp[(size_t)ka * NPT + nb + l16];
    float f1 = xp[(size_t)(ka + 1) * NPT + nb + l16];
    if (clampX) { f0 = fmaxf(f0, EPSF); f1 = fmaxf(f1, EPSF); }
    a.x = f0; a.y = f1;
    const float* r0 = bp + (size_t)ka * RR + l16;
    const float* r1 = r0 + RR;
    v2f b0, b1, b2, b3;
    b0.x = r0[0];  b0.y = r1[0];
    b1.x = r0[16]; b1.y = r1[16];
    b2.x = r0[32]; b2.y = r1[32];
    b3.x = r0[48]; b3.y = r1[48];
    a0 = wmma4(a, b0, a0); a1 = wmma4(a, b1, a1);
    a2 = wmma4(a, b2, a2); a3 = wmma4(a, b3, a3);
  }
  float* op = numC + (size_t)bs * NPT * RR;
  #pragma unroll
  for (int i = 0; i < 8; i++) {
    size_t row = (size_t)(nb + i + 8 * half) * RR;
    op[row + l16]      = a0[i];
    op[row + 16 + l16] = a1[i];
    op[row + 32 + l16] = a2[i];
    op[row + 48 + l16] = a3[i];
  }
}

// BtB[bs][r][s] = sum_d Bn[d][r]*Bn[d][s]; block = 512 (16 waves = 16 tiles); grid (BS)
__global__ void k_btb(const float* __restrict__ Bn, float* __restrict__ BtB) {
  int bs   = blockIdx.x;
  int lane = threadIdx.x & 31, wave = threadIdx.x >> 5;
  int l16  = lane & 15, half = lane >> 4;
  int rt   = wave >> 2, st = wave & 3;
  const float* bp = Bn + (size_t)bs * DD * RR;
  v8f acc = {};
  for (int k = 0; k < DD; k += 4) {
    int ka = k + 2 * half;
    v2f a, b;
    a.x = bp[(size_t)ka * RR + rt * 16 + l16];
    a.y = bp[(size_t)(ka + 1) * RR + rt * 16 + l16];
    b.x = bp[(size_t)ka * RR + st * 16 + l16];
    b.y = bp[(size_t)(ka + 1) * RR + st * 16 + l16];
    acc = wmma4(a, b, acc);
  }
  float* o = BtB + (size_t)bs * RR * RR;
  #pragma unroll
  for (int i = 0; i < 8; i++)
    o[(rt * 16 + i + 8 * half) * RR + st * 16 + l16] = acc[i];
}

// c <- c * numC / (c*BtB + eps), in place; wave -> 16 n x 64 r; grid (N/64, BS)
__global__ void k_update_c(float* __restrict__ c, const float* __restrict__ numC,
                           const float* __restrict__ BtB) {
  int bs   = blockIdx.y;
  int lane = threadIdx.x & 31, wave = threadIdx.x >> 5;
  int l16  = lane & 15, half = lane >> 4;
  int nb   = blockIdx.x * 64 + wave * 16;
  float* cp = c + (size_t)bs * NPT * RR;
  const float* bb = BtB + (size_t)bs * RR * RR;
  v8f d0 = {}, d1 = {}, d2 = {}, d3 = {};
  for (int k = 0; k < RR; k += 4) {
    int ka = k + 2 * half;
    v2f a;
    a.x = cp[(size_t)(nb + l16) * RR + ka];
    a.y = cp[(size_t)(nb + l16) * RR + ka + 1];
    const float* r0 = bb + ka * RR + l16;
    const float* r1 = r0 + RR;
    v2f b0, b1, b2, b3;
    b0.x = r0[0];  b0.y = r1[0];
    b1.x = r0[16]; b1.y = r1[16];
    b2.x = r0[32]; b2.y = r1[32];
    b3.x = r0[48]; b3.y = r1[48];
    d0 = wmma4(a, b0, d0); d1 = wmma4(a, b1, d1);
    d2 = wmma4(a, b2, d2); d3 = wmma4(a, b3, d3);
  }
  const float* np = numC + (size_t)bs * NPT * RR;
  #pragma unroll
  for (int i = 0; i < 8; i++) {
    size_t row = (size_t)(nb + i + 8 * half) * RR;
    size_t i0 = row + l16, i1 = row + 16 + l16, i2 = row + 32 + l16, i3 = row + 48 + l16;
    cp[i0] = cp[i0] * np[i0] / (d0[i] + EPSF);
    cp[i1] = cp[i1] * np[i1] / (d1[i] + EPSF);
    cp[i2] = cp[i2] * np[i2] / (d2[i] + EPSF);
    cp[i3] = cp[i3] * np[i3] / (d3[i] + EPSF);
  }
}

// CtC[bs][r][s] += sum over chunk of n; split-K with atomics; block 512 = 16 tiles; grid (N/2048, BS)
__global__ void k_ctc(const float* __restrict__ c, float* __restrict__ CtC) {
  int bs   = blockIdx.y;
  int lane = threadIdx.x & 31, wave = threadIdx.x >> 5;
  int l16  = lane & 15, half = lane >> 4;
  int rt   = wave >> 2, st = wave & 3;
  int kb   = blockIdx.x * 2048;
  const float* cp = c + (size_t)bs * NPT * RR;
  v8f acc = {};
  for (int k = 0; k < 2048; k += 4) {
    int ka = kb + k + 2 * half;
    v2f a, b;
    a.x = cp[(size_t)ka * RR + rt * 16 + l16];
    a.y = cp[(size_t)(ka + 1) * RR + rt * 16 + l16];
    b.x = cp[(size_t)ka * RR + st * 16 + l16];
    b.y = cp[(size_t)(ka + 1) * RR + st * 16 + l16];
    acc = wmma4(a, b, acc);
  }
  float* o = CtC + (size_t)bs * RR * RR;
  #pragma unroll
  for (int i = 0; i < 8; i++)
    atomicAdd(&o[(rt * 16 + i + 8 * half) * RR + st * 16 + l16], acc[i]);
}

// numB[bs][d][r] += sum over chunk of n of xc[d][n]*c[n][r]; block 256 (8 waves, each 64 d);
// grid (N/1024, BS)
__global__ void k_numb(const float* __restrict__ x, const float* __restrict__ c,
                       float* __restrict__ numB) {
  int bs   = blockIdx.y;
  int lane = threadIdx.x & 31, wave = threadIdx.x >> 5;
  int l16  = lane & 15, half = lane >> 4;
  int kb   = blockIdx.x * 1024;
  int db   = wave * 64;
  const float* xp = x + (size_t)bs * DD * NPT;
  const float* cp = c + (size_t)bs * NPT * RR;
  v8f acc[4][4] = {};
  for (int k = 0; k < 1024; k += 4) {
    int ka = kb + k + 2 * half;
    const float* r0 = cp + (size_t)ka * RR + l16;
    const float* r1 = r0 + RR;
    v2f b0, b1, b2, b3;
    b0.x = r0[0];  b0.y = r1[0];
    b1.x = r0[16]; b1.y = r1[16];
    b2.x = r0[32]; b2.y = r1[32];
    b3.x = r0[48]; b3.y = r1[48];
    #pragma unroll
    for (int mt = 0; mt < 4; mt++) {
      size_t xrow = (size_t)(db + mt * 16 + l16) * NPT;
      v2f a;
      a.x = fmaxf(xp[xrow + ka], EPSF);
      a.y = fmaxf(xp[xrow + ka + 1], EPSF);
      acc[mt][0] = wmma4(a, b0, acc[mt][0]);
      acc[mt][1] = wmma4(a, b1, acc[mt][1]);
      acc[mt][2] = wmma4(a, b2, acc[mt][2]);
      acc[mt][3] = wmma4(a, b3, acc[mt][3]);
    }
  }
  float* o = numB + (size_t)bs * DD * RR;
  #pragma unroll
  for (int mt = 0; mt < 4; mt++)
    #pragma unroll
    for (int rt = 0; rt < 4; rt++)
      #pragma unroll
      for (int i = 0; i < 8; i++)
        atomicAdd(&o[(db + mt * 16 + i + 8 * half) * RR + rt * 16 + l16], acc[mt][rt][i]);
}

// out[bs][d][n] = sum_r Bn[d][r]*c[n][r]; wave -> 16 d x 64 n; grid (N/64, D/64, BS); block 128
__global__ void k_xhat(const float* __restrict__ Bn, const float* __restrict__ c,
                       float* __restrict__ out) {
  int bs   = blockIdx.z;
  int lane = threadIdx.x & 31, wave = threadIdx.x >> 5;
  int l16  = lane & 15, half = lane >> 4;
  int db   = blockIdx.y * 64 + wave * 16;
  int nb   = blockIdx.x * 64;
  const float* bp = Bn + (size_t)bs * DD * RR;
  const float* cp = c + (size_t)bs * NPT * RR;
  v8f a0 = {}, a1 = {}, a2 = {}, a3 = {};
  for (int k = 0; k < RR; k += 4) {
    int ka = k + 2 * half;
    v2f a;
    a.x = bp[(size_t)(db + l16) * RR + ka];
    a.y = bp[(size_t)(db + l16) * RR + ka + 1];
    v2f b0, b1, b2, b3;
    b0.x = cp[(size_t)(nb + l16) * RR + ka];      b0.y = cp[(size_t)(nb + l16) * RR + ka + 1];
    b1.x = cp[(size_t)(nb + 16 + l16) * RR + ka]; b1.y = cp[(size_t)(nb + 16 + l16) * RR + ka + 1];
    b2.x = cp[(size_t)(nb + 32 + l16) * RR + ka]; b2.y = cp[(size_t)(nb + 32 + l16) * RR + ka + 1];
    b3.x = cp[(size_t)(nb + 48 + l16) * RR + ka]; b3.y = cp[(size_t)(nb + 48 + l16) * RR + ka + 1];
    a0 = wmma4(a, b0, a0); a1 = wmma4(a, b1, a1);
    a2 = wmma4(a, b2, a2); a3 = wmma4(a, b3, a3);
  }
  float* op = out + (size_t)bs * DD * NPT;
  #pragma unroll
  for (int i = 0; i < 8; i++) {
    size_t row = (size_t)(db + i + 8 * half) * NPT + nb;
    op[row + l16]      = a0[i];
    op[row + 16 + l16] = a1[i];
    op[row + 32 + l16] = a2[i];
    op[row + 48 + l16] = a3[i];
  }
}

// ---------------- host orchestration ----------------

extern "C" void kernel_launch(void* const* d_in, const int* in_sizes, int n_in,
                              void* d_out, int out_size, void* d_ws, size_t ws_size,
                              hipStream_t stream) {
  (void)in_sizes; (void)n_in; (void)out_size; (void)ws_size;
  const float* x     = (const float*)d_in[0];   // (2, 512, 16, 64, 64) = (2, 512, 65536)
  const float* bases = (const float*)d_in[1];   // (2, 512, 64)
  float* out = (float*)d_out;

  float* W = (float*)d_ws;
  float* Bn   = W;                  // 65536
  float* Btmp = W + 65536;          // 65536
  float* BtB  = W + 131072;         // 8192
  float* CtC  = W + 139264;         // 8192
  float* numB = W + 147456;         // 65536
  float* numC = W + 212992;         // 8388608
  float* cbuf = W + 8601600;        // 8388608

  dim3 gStrip(NPT / 64, BSN);       // per-wave 16xN strip kernels, block 128

  // init: b = l2norm(bases); coef = softmax(x^T b)  (unclamped x)
  k_norm_bases<<<dim3(BSN), 256, 0, stream>>>(bases, Bn);
  k_gemm_xtb<<<gStrip, 128, 0, stream>>>(x, Bn, numC, 0);
  k_softmax<<<dim3(NPT / 256, BSN), 256, 0, stream>>>(numC, cbuf);

  for (int step = 0; step < 6; step++) {
    k_btb<<<dim3(BSN), 512, 0, stream>>>(Bn, BtB);
    k_gemm_xtb<<<gStrip, 128, 0, stream>>>(x, Bn, numC, 1);
    k_update_c<<<gStrip, 128, 0, stream>>>(cbuf, numC, BtB);
    k_zero<<<dim3((BSN * RR * RR + 255) / 256), 256, 0, stream>>>(CtC, BSN * RR * RR);
    k_ctc<<<dim3(NPT / 2048, BSN), 512, 0, stream>>>(cbuf, CtC);
    k_zero<<<dim3((BSN * DD * RR + 255) / 256), 256, 0, stream>>>(numB, BSN * DD * RR);
    k_numb<<<dim3(NPT / 1024, BSN), 256, 0, stream>>>(x, cbuf, numB);
    k_update_b<<<dim3(BSN), 256, 0, stream>>>(Bn, numB, CtC, Btmp);
    k_clamp<<<dim3(4096), 256, 0, stream>>>(cbuf, BSN * NPT * RR);
  }

  // compute_coef + reconstruction
  k_btb<<<dim3(BSN), 512, 0, stream>>>(Bn, BtB);
  k_gemm_xtb<<<gStrip, 128, 0, stream>>>(x, Bn, numC, 1);
  k_update_c<<<gStrip, 128, 0, stream>>>(cbuf, numC, BtB);
  k_xhat<<<dim3(NPT / 64, DD / 64, BSN), 128, 0, stream>>>(Bn, cbuf, out);
}